// Pe_i_CrossAttention_5987184410794
// MI455X (gfx1250) — compile-verified
//
#include <hip/hip_runtime.h>

// Problem constants (from reference setup)
#define NPTS 16384      // N
#define NB_K 16         // K neighbors
#define CCH  64         // C
#define CIN  67         // C + 3
#define CPAD 68         // padded K-dim for WMMA (17 steps of 4)
#define WAVES 4         // waves per workgroup (block = 128 threads)
#define WROW 288        // LDS row stride (floats) per channel-pair: 288%64=32
                        // -> half0/half1 pair-rows land in disjoint bank halves

typedef __attribute__((ext_vector_type(2))) float v2f;
typedef __attribute__((ext_vector_type(8))) float v8f;

__device__ __forceinline__ v8f wmma4(v2f a, v2f b, v8f c) {
  // D = A(16x4 f32) * B(4x16 f32) + C(16x16 f32)
  return __builtin_amdgcn_wmma_f32_16x16x4_f32(false, a, false, b, (short)0, c,
                                               false, false);
}

// butterfly reductions across the 16 lanes of each half-wave (xor masks 1,2,4,8)
__device__ __forceinline__ float half_max(float v) {
  v = fmaxf(v, __shfl_xor(v, 1, 32));
  v = fmaxf(v, __shfl_xor(v, 2, 32));
  v = fmaxf(v, __shfl_xor(v, 4, 32));
  v = fmaxf(v, __shfl_xor(v, 8, 32));
  return v;
}
__device__ __forceinline__ float half_sum(float v) {
  v += __shfl_xor(v, 1, 32);
  v += __shfl_xor(v, 2, 32);
  v += __shfl_xor(v, 4, 32);
  v += __shfl_xor(v, 8, 32);
  return v;
}

// Per-point epilogue: energies from K-accumulators + LDS q, softmax over the
// 16 neighbors (lanes of the half), attn-weighted sum of V via a fold-tree
// reduction that leaves lane j holding channel (16m + (j&7) + 8*half).
// acc[0..3] = K tiles (channels 0..63), acc[4..7] = V tiles.
__device__ __forceinline__ void attend_store(const v8f* acc, const float* qrow,
                                             float* outp, int kk, int half) {
  float x[4];
#pragma unroll
  for (int m = 0; m < 4; ++m) {
    const float* qp = qrow + 16 * m + 8 * half;   // q channels, head 2m+half
    float e = 0.f;
#pragma unroll
    for (int r = 0; r < 8; ++r) e = fmaf(qp[r], acc[m][r], e);
    e *= 0.35355339059327373f;                    // 1/sqrt(D), D=8
    float emax = half_max(e);
    float pe = __expf(e - emax);
    float ps = half_sum(pe);
    float attn = pe / ps;                         // this lane's attn for k=kk

    float t[8];
#pragma unroll
    for (int r = 0; r < 8; ++r) t[r] = attn * acc[4 + m][r];
    // fold-tree: reduce 8 regs across 16 lanes; end: lane holds sum for r=kk&7
#pragma unroll
    for (int r = 0; r < 8; ++r) t[r] += __shfl_xor(t[r], 1, 32);
    float u[4];
#pragma unroll
    for (int q = 0; q < 4; ++q) u[q] = (kk & 1) ? t[2 * q + 1] : t[2 * q];
#pragma unroll
    for (int q = 0; q < 4; ++q) u[q] += __shfl_xor(u[q], 2, 32);
    float w0 = (kk & 2) ? u[1] : u[0];
    float w1 = (kk & 2) ? u[3] : u[2];
    w0 += __shfl_xor(w0, 4, 32);
    w1 += __shfl_xor(w1, 4, 32);
    float xx = (kk & 4) ? w1 : w0;
    xx += __shfl_xor(xx, 8, 32);
    x[m] = xx;                                    // = S[m][kk&7], all-lane valid
  }
  // lanes kk<8 own tiles {0,1}, lanes kk>=8 own tiles {2,3}; 2 stores per lane
  float va = (kk & 8) ? x[2] : x[0];
  float vb = (kk & 8) ? x[3] : x[1];
  int cA = ((kk & 8) ? 32 : 0) + (kk & 7) + 8 * half;
  outp[(size_t)cA * NPTS] = va;
  outp[(size_t)(cA + 16) * NPTS] = vb;
}

__global__ __launch_bounds__(WAVES * 32, 1)
void pct_attn_kernel(const float* __restrict__ pcd,
                     const float* __restrict__ neighbors,
                     const float* __restrict__ xyz,
                     const float* __restrict__ Wq,
                     const float* __restrict__ Wk,
                     const float* __restrict__ Wv,
                     const int* __restrict__ idx_all,
                     float* __restrict__ out) {
  // [Wk|Wv] stored as channel pairs: pair p=(c>>1) holds channels (2p,2p+1)
  // interleaved per column -> A-fragment = one aligned ds_load_b64.
  __shared__ float lds_wkv[34 * WROW];
  __shared__ float lds_q[WAVES * 16 * CCH];    // per-wave Q tile [p][c]

  const int tid = threadIdx.x;
  for (int i = tid; i < 34 * 256; i += WAVES * 32) {
    int pair = i >> 8;            // 0..33
    int col2 = i & 255;           // col*2 + odd
    int col = col2 >> 1, odd = col2 & 1;
    int c = 2 * pair + odd;
    float v = 0.f;
    if (c < CIN)
      v = (col < 64) ? Wk[col * CIN + c] : Wv[(col - 64) * CIN + c];
    lds_wkv[pair * WROW + col2] = v;
  }
  __syncthreads();

  const int lane = tid & 31;
  const int wv   = tid >> 5;
  const int half = lane >> 4;
  const int kk   = lane & 15;

  const int tile = blockIdx.x * WAVES + wv;    // 4096 tiles of 16 points
  const int b  = tile >> 10;                   // 1024 tiles per batch
  const int n0 = (tile & 1023) << 4;

  float* qs = lds_q + wv * (16 * CCH);

  // ---------------- Q phase: Q(64 x 16pts) = Wq(64x67) * pcd_cat(67x16) ----
  {
    const float* pb = pcd + (size_t)b * CCH * NPTS + n0;
    const float* xb = xyz + (size_t)b * 3 * NPTS + n0;
    const int voff = half * 2 * NPTS + kk;     // B-frag lane offset
    float2 bq[17];
#pragma unroll
    for (int s = 0; s < 16; ++s) {
      bq[s].x = pb[(size_t)(4 * s    ) * NPTS + voff];
      bq[s].y = pb[(size_t)(4 * s + 1) * NPTS + voff];
    }
    bq[16].x = xb[(half ? 2 : 0) * NPTS + kk];
    bq[16].y = half ? 0.f : xb[NPTS + kk];

#pragma unroll
    for (int m = 0; m < 4; ++m) {
      v8f acc = {0, 0, 0, 0, 0, 0, 0, 0};
#pragma unroll
      for (int s = 0; s <= 16; ++s) {
        int c0 = 4 * s + 2 * half;                    // <= 66, always valid
        int c1 = (c0 + 1 < CIN) ? c0 + 1 : CIN - 1;   // clamp (only s=16,h=1)
        int row = 16 * m + kk;
        float a0 = Wq[row * CIN + c0];
        float a1 = Wq[row * CIN + c1];
        a1 = (c0 + 1 < CIN) ? a1 : 0.f;               // branchless pad
        v2f A; A[0] = a0; A[1] = a1;
        v2f B; B[0] = bq[s].x; B[1] = bq[s].y;
        acc = wmma4(A, B, acc);
      }
#pragma unroll
      for (int r = 0; r < 8; ++r)               // store Q transposed: [p][c]
        qs[kk * CCH + 16 * m + 8 * half + r] = acc[r];
    }
  }

  // ---------------- KV + attention, two points per iteration --------------
  const size_t NK = (size_t)NPTS * NB_K;
  const float* nbB = neighbors + (size_t)b * CCH * NK;
  const float* xb2 = xyz + (size_t)b * 3 * NPTS;
  const int* idxB  = idx_all + (size_t)b * NPTS * NB_K;
  float* outB = out + (size_t)b * CCH * NPTS + n0;
  const size_t voffn = (size_t)(half * 2) * NK + kk;

  for (int it = 0; it < 8; ++it) {
    const int p0 = 2 * it, p1 = 2 * it + 1;
    const float* nb0 = nbB + (size_t)(n0 + p0) * NB_K;
    const float* nb1 = nb0 + NB_K;

    float2 b0[17], b1[17];
#pragma unroll
    for (int s = 0; s < 16; ++s) {              // channels 0..63 from neighbors
      b0[s].x = nb0[(size_t)(4 * s    ) * NK + voffn];
      b0[s].y = nb0[(size_t)(4 * s + 1) * NK + voffn];
      b1[s].x = nb1[(size_t)(4 * s    ) * NK + voffn];
      b1[s].y = nb1[(size_t)(4 * s + 1) * NK + voffn];
    }
    {                                           // channels 64..66: xyz gather
      const int* ib = idxB + (size_t)(n0 + p0) * NB_K;
      int i0 = ib[kk];
      int i1 = ib[NB_K + kk];
      int drow = (half ? 2 : 0) * NPTS;
      b0[16].x = xb2[drow + i0];
      b0[16].y = half ? 0.f : xb2[NPTS + i0];
      b1[16].x = xb2[drow + i1];
      b1[16].y = half ? 0.f : xb2[NPTS + i1];
    }

    v8f acc0[8], acc1[8];
#pragma unroll
    for (int m = 0; m < 8; ++m) {
      acc0[m] = (v8f){0, 0, 0, 0, 0, 0, 0, 0};
      acc1[m] = (v8f){0, 0, 0, 0, 0, 0, 0, 0};
    }
#pragma unroll
    for (int m = 0; m < 8; ++m) {               // m 0..3 = Wk rows, 4..7 = Wv
#pragma unroll
      for (int s = 0; s <= 16; ++s) {
        // one ds_load_b64: channels (4s+2h, 4s+2h+1) of output column 16m+kk
        const float2 A2 =
            *(const float2*)(lds_wkv + (2 * s + half) * WROW + (16 * m + kk) * 2);
        v2f A;  A[0] = A2.x;     A[1] = A2.y;
        v2f B0; B0[0] = b0[s].x; B0[1] = b0[s].y;
        v2f B1; B1[0] = b1[s].x; B1[1] = b1[s].y;
        acc0[m] = wmma4(A, B0, acc0[m]);        // A shared by both points
        acc1[m] = wmma4(A, B1, acc1[m]);
      }
    }
    attend_store(acc0, qs + p0 * CCH, outB + p0, kk, half);
    attend_store(acc1, qs + p1 * CCH, outB + p1, kk, half);
  }
}

extern "C" void kernel_launch(void* const* d_in, const int* in_sizes, int n_in,
                              void* d_out, int out_size, void* d_ws,
                              size_t ws_size, hipStream_t stream) {
  const float* pcd       = (const float*)d_in[0];
  const float* neighbors = (const float*)d_in[1];
  const float* xyz       = (const float*)d_in[2];
  const float* Wq        = (const float*)d_in[3];
  const float* Wk        = (const float*)d_in[4];
  const float* Wv        = (const float*)d_in[5];
  const int*   idx_all   = (const int*)d_in[6];
  float* out = (float*)d_out;

  // 4 batches * (16384/16) tiles = 4096 wave-tiles; 4 waves per block.
  dim3 grid(1024), block(WAVES * 32);
  hipLaunchKernelGGL(pct_attn_kernel, grid, block, 0, stream,
                     pcd, neighbors, xyz, Wq, Wk, Wv, idx_all, out);
}